// LSTMModel_79061757984946
// MI455X (gfx1250) — compile-verified
//
#include <hip/hip_runtime.h>
#include <math.h>
#include <stdint.h>

// Problem constants (from reference)
#define TSEQ 1024
#define DIN  256
#define HF   256
#define G4   1024   // 4*H
#define HIDN 512
#define OUTN 256

typedef __attribute__((ext_vector_type(2))) float v2f;
typedef __attribute__((ext_vector_type(8))) float v8f;
typedef __attribute__((ext_vector_type(4))) unsigned int v4u;
typedef __attribute__((ext_vector_type(8))) int v8i;
typedef __attribute__((ext_vector_type(4))) int v4i;

// LDS panel strides (floats) chosen for conflict-free WMMA fragment reads.
// A: rows of 256 dwords padded by 4  -> stride 260 (pad_interval=7, pad_amount=3)
// B: rows of 64 dwords padded by 8   -> stride 72  (pad_interval=5, pad_amount=7)
#define AS_STRIDE 260
#define BS_STRIDE 72
#define AS_FLOATS (64 * AS_STRIDE)
#define BS_FLOATS (256 * BS_STRIDE)
#define GEMM_LDS_BYTES ((AS_FLOATS + BS_FLOATS) * 4)

// ---------------------------------------------------------------------------
// Tensor Data Mover: 2D tile (global -> LDS) with optional LDS padding.
// D# built per CDNA5 ISA (group0 128b + group1 256b,
// groups 2/3 zero => 2D tensor, tile_dim2=0 terminates the dims).
// ---------------------------------------------------------------------------
__device__ static inline void tdm_load_2d(uint32_t lds_off, const void* gptr,
                                          uint32_t tensor_d0, uint32_t tensor_d1,
                                          uint64_t stride0_elems,
                                          uint32_t tile_d0, uint32_t tile_d1,
                                          uint32_t pad_interval, uint32_t pad_amount) {
  const uint64_t ga = (uint64_t)(uintptr_t)gptr;
  v4u g0;
  g0.x = 1u;                                               // count=1, user descriptor
  g0.y = lds_off;                                          // lds_addr [63:32]
  g0.z = (uint32_t)ga;                                     // global_addr [95:64]
  g0.w = ((uint32_t)(ga >> 32) & 0x1FFFFFFu) | (2u << 30); // addr[56:32] | type=2

  const uint32_t d0 = (2u << 16)                 // data_size = 4 bytes
                    | (1u << 20)                 // pad_enable
                    | (pad_interval << 22)       // pad_interval [24:22]
                    | (pad_amount << 25);        // pad_amount  [31:25]
  const uint32_t d1 = (tensor_d0 & 0xFFFFu) << 16;                      // dim0[15:0]
  const uint32_t d2 = (tensor_d0 >> 16) | ((tensor_d1 & 0xFFFFu) << 16);// dim0 hi | dim1 lo
  const uint32_t d3 = (tensor_d1 >> 16) | (tile_d0 << 16);              // dim1 hi | tile0
  const uint32_t d4 = tile_d1;                                          // tile1 | tile2=0
  const uint32_t d5 = (uint32_t)(stride0_elems & 0xFFFFFFFFu);          // stride0 lo
  const uint32_t d6 = (uint32_t)(stride0_elems >> 32) & 0xFFFFu;        // stride0 hi
  const v8i g1 = {(int)d0, (int)d1, (int)d2, (int)d3,
                  (int)d4, (int)d5, (int)d6, 0};
  const v4i z4 = {0, 0, 0, 0};
#if __has_include(<hip/amd_detail/amd_gfx1250_TDM.h>)
  const v8i z8 = {0, 0, 0, 0, 0, 0, 0, 0};
  __builtin_amdgcn_tensor_load_to_lds(g0, g1, z4, z4, z8, 0);
#else
  __builtin_amdgcn_tensor_load_to_lds(g0, g1, z4, z4, 0);
#endif
}

// ---------------------------------------------------------------------------
// C[1024,1024] = A[1024,256] @ B[256,1024], f32, V_WMMA_F32_16X16X4_F32.
// Block = 256 threads (8 waves) computes a 64x64 C tile. Whole K=256 A/B
// panels are TDM-DMA'd into LDS once (with bank-conflict padding), then each
// wave runs 64 WMMA K-steps on a 16x32 strip (A fragment reused for 2 tiles).
// ---------------------------------------------------------------------------
__global__ __launch_bounds__(256) void gemm_1024x1024x256_f32wmma(
    const float* __restrict__ A, const float* __restrict__ B,
    float* __restrict__ C) {
  constexpr int N = 1024, K = 256;
  extern __shared__ float smem[];
  float* As = smem;               // [64][AS_STRIDE]
  float* Bs = smem + AS_FLOATS;   // [256][BS_STRIDE]

  const int wave = threadIdx.x >> 5;
  const int lane = threadIdx.x & 31;
  const int bm = blockIdx.y;      // 16 tiles along M
  const int bn = blockIdx.x;      // 16 tiles along N

  if (wave == 0) {
    const uint32_t lds_base = (uint32_t)(uintptr_t)smem;
    // A panel: 64 rows x 256 cols, tile start at A[bm*64][0]
    tdm_load_2d(lds_base, A + (size_t)bm * 64 * K,
                /*tensor_d0=*/K, /*tensor_d1=*/64, /*stride0=*/K,
                /*tile_d0=*/K, /*tile_d1=*/64,
                /*pad_interval=*/7, /*pad_amount=*/3);       // +4 dw / 256 dw
    // B panel: 256 rows x 64 cols, tile start at B[0][bn*64]
    tdm_load_2d(lds_base + AS_FLOATS * 4, B + bn * 64,
                /*tensor_d0=*/N, /*tensor_d1=*/K, /*stride0=*/N,
                /*tile_d0=*/64, /*tile_d1=*/K,
                /*pad_interval=*/5, /*pad_amount=*/7);       // +8 dw / 64 dw
    __builtin_amdgcn_s_wait_tensorcnt(0);
  }
  __syncthreads();

  // Wave -> 16(M) x 32(N) strip: wm in 0..3, wn in 0..1 (two 16x16 tiles).
  const int wm = wave >> 1;
  const int wn = wave & 1;
  const int mloc = (wm << 4) + (lane & 15);
  const int n0 = (wn << 5) + (lane & 15);
  const int n1 = n0 + 16;
  const int khalf = (lane >> 4) << 1;      // 0 | 2

  v8f acc0 = {}, acc1 = {};
  const float* arow = As + mloc * AS_STRIDE;
#pragma unroll 4
  for (int k0 = 0; k0 < K; k0 += 4) {
    const int ka = k0 + khalf;
    v2f a = *(const v2f*)(arow + ka);
    v2f b0, b1;
    b0.x = Bs[ka * BS_STRIDE + n0];
    b0.y = Bs[(ka + 1) * BS_STRIDE + n0];
    b1.x = Bs[ka * BS_STRIDE + n1];
    b1.y = Bs[(ka + 1) * BS_STRIDE + n1];
    acc0 = __builtin_amdgcn_wmma_f32_16x16x4_f32(false, a, false, b0,
                                                 (short)0, acc0, false, false);
    acc1 = __builtin_amdgcn_wmma_f32_16x16x4_f32(false, a, false, b1,
                                                 (short)0, acc1, false, false);
  }

  const int crow = bm * 64 + (wm << 4) + ((lane >> 4) << 3);
  const int col0 = bn * 64 + n0;
  const int col1 = bn * 64 + n1;
#pragma unroll
  for (int g = 0; g < 8; ++g) {
    C[(size_t)(crow + g) * N + col0] = acc0[g];
    C[(size_t)(crow + g) * N + col1] = acc1[g];
  }
}

// ---------------------------------------------------------------------------
// Sequential LSTM scan: one persistent workgroup, 256 threads (8 waves).
// h lives in LDS; Wh (1 MB) streams from the 192 MB L2 every step.
// ---------------------------------------------------------------------------
__global__ __launch_bounds__(256) void lstm_scan(
    const float* __restrict__ xz,   // [TSEQ, 4H] precomputed x@Wx
    const float* __restrict__ Wh,   // [H, 4H]
    const float* __restrict__ bias, // [4H]
    float* __restrict__ hs) {       // [TSEQ, H]
  __shared__ float hbuf[HF];
  const int j = threadIdx.x;
  float c = 0.0f;
  hbuf[j] = 0.0f;
  __syncthreads();

  const float bi = bias[j];
  const float bf = bias[HF + j];
  const float bg = bias[2 * HF + j];
  const float bo = bias[3 * HF + j];

  for (int t = 0; t < TSEQ; ++t) {
    const float* xzt = xz + (size_t)t * G4;
    float zi = xzt[j] + bi;
    float zf = xzt[HF + j] + bf;
    float zg = xzt[2 * HF + j] + bg;
    float zo = xzt[3 * HF + j] + bo;

#pragma unroll 8
    for (int k = 0; k < HF; ++k) {
      const float hk = hbuf[k];
      const float* w = Wh + (size_t)k * G4;
      zi = fmaf(hk, w[j], zi);
      zf = fmaf(hk, w[HF + j], zf);
      zg = fmaf(hk, w[2 * HF + j], zg);
      zo = fmaf(hk, w[3 * HF + j], zo);
    }

    const float ig = 1.0f / (1.0f + __expf(-zi));
    const float fg = 1.0f / (1.0f + __expf(-zf));
    const float gg = tanhf(zg);
    const float og = 1.0f / (1.0f + __expf(-zo));
    c = fmaf(fg, c, ig * gg);
    const float h = og * tanhf(c);

    __syncthreads();              // everyone done reading old h
    hbuf[j] = h;
    hs[(size_t)t * HF + j] = h;
    __syncthreads();              // new h visible before next step
  }
}

// ---------------------------------------------------------------------------
// Dense head part 1: split-K matvec over Wd1 (537 MB stream, HBM-bound),
// deterministic partials (no float atomics -> bit-identical graph replays).
// ---------------------------------------------------------------------------
__global__ __launch_bounds__(256) void dense1_partial(
    const float* __restrict__ flat, const float* __restrict__ Wd1,
    float* __restrict__ part) {     // [256 blocks, 512]
  const int t = threadIdx.x;
  const size_t kbase = (size_t)blockIdx.x * 1024;
  float acc0 = 0.0f, acc1 = 0.0f;
  for (int k = 0; k < 1024; ++k) {
    const float v = flat[kbase + k];                 // uniform -> scalar load
    const float* w = Wd1 + (kbase + k) * HIDN;
    __builtin_prefetch(w + 8 * HIDN, 0, 1);          // global_prefetch_b8
    acc0 = fmaf(v, w[t], acc0);
    acc1 = fmaf(v, w[t + 256], acc1);
  }
  part[(size_t)blockIdx.x * HIDN + t] = acc0;
  part[(size_t)blockIdx.x * HIDN + t + 256] = acc1;
}

__global__ __launch_bounds__(256) void dense1_reduce(
    const float* __restrict__ part, float* __restrict__ y1) {
  const int c = blockIdx.x * 256 + threadIdx.x;      // 0..511
  float s = 0.0f;
  for (int b = 0; b < 256; ++b) s += part[(size_t)b * HIDN + c];
  y1[c] = s;
}

// ---------------------------------------------------------------------------
// Dense head part 2: out[256] = relu(y1 + bd1) @ Wd2[512,256] + bd2
// ---------------------------------------------------------------------------
__global__ __launch_bounds__(256) void dense2(
    const float* __restrict__ y1, const float* __restrict__ bd1,
    const float* __restrict__ Wd2, const float* __restrict__ bd2,
    float* __restrict__ out) {
  __shared__ float s[HIDN];
  const int t = threadIdx.x;
  s[t] = fmaxf(y1[t] + bd1[t], 0.0f);
  s[t + 256] = fmaxf(y1[t + 256] + bd1[t + 256], 0.0f);
  __syncthreads();
  float acc = bd2[t];
#pragma unroll 4
  for (int h = 0; h < HIDN; ++h) acc = fmaf(s[h], Wd2[(size_t)h * OUTN + t], acc);
  out[t] = acc;
}

// ---------------------------------------------------------------------------
extern "C" void kernel_launch(void* const* d_in, const int* in_sizes, int n_in,
                              void* d_out, int out_size, void* d_ws, size_t ws_size,
                              hipStream_t stream) {
  const float* x   = (const float*)d_in[0];
  const float* Wx1 = (const float*)d_in[1];
  const float* Wh1 = (const float*)d_in[2];
  const float* b1  = (const float*)d_in[3];
  const float* Wx2 = (const float*)d_in[4];
  const float* Wh2 = (const float*)d_in[5];
  const float* b2  = (const float*)d_in[6];
  const float* Wd1 = (const float*)d_in[7];
  const float* bd1 = (const float*)d_in[8];
  const float* Wd2 = (const float*)d_in[9];
  const float* bd2 = (const float*)d_in[10];
  float* out = (float*)d_out;

  // Workspace layout (floats): XZ reused between the two layers.
  float* ws   = (float*)d_ws;
  float* XZ   = ws;                       // 1024*1024      (4 MB)
  float* HS1  = XZ + 1024 * 1024;         // 1024*256       (1 MB)
  float* HS2  = HS1 + TSEQ * HF;          // 1024*256       (1 MB)
  float* Y1   = HS2 + TSEQ * HF;          // 512
  float* PART = Y1 + HIDN;                // 256*512        (0.5 MB)

  dim3 ggrid(16, 16);

  // Layer 1: input projection (TDM + WMMA) + scan
  gemm_1024x1024x256_f32wmma<<<ggrid, 256, GEMM_LDS_BYTES, stream>>>(x, Wx1, XZ);
  lstm_scan<<<1, 256, 0, stream>>>(XZ, Wh1, b1, HS1);
  // Layer 2
  gemm_1024x1024x256_f32wmma<<<ggrid, 256, GEMM_LDS_BYTES, stream>>>(HS1, Wx2, XZ);
  lstm_scan<<<1, 256, 0, stream>>>(XZ, Wh2, b2, HS2);
  // Dense head
  dense1_partial<<<256, 256, 0, stream>>>(HS2, Wd1, PART);
  dense1_reduce<<<2, 256, 0, stream>>>(PART, Y1);
  dense2<<<1, 256, 0, stream>>>(Y1, bd1, Wd2, bd2, out);
}